// GraphNeuralNetwork_83554293776709
// MI455X (gfx1250) — compile-verified
//
#include <hip/hip_runtime.h>
#include <hip/hip_bf16.h>
#include <math.h>

typedef float v2f __attribute__((ext_vector_type(2)));
typedef float v8f __attribute__((ext_vector_type(8)));

#define DIN 256
#define HID 128
#define NC  8

// ---------------- degree / normalization ----------------

__global__ void k_deg_init(float* deg, int n) {
    int i = blockIdx.x * blockDim.x + threadIdx.x;
    if (i < n) deg[i] = 1.0f;   // self-loop contributes 1 to every dst
}

__global__ void k_deg_count(const long long* __restrict__ dst, float* deg, int e) {
    int i = blockIdx.x * blockDim.x + threadIdx.x;
    if (i < e) atomicAdd(&deg[(int)dst[i]], 1.0f);
}

__global__ void k_dinv(float* deg, int n) {
    int i = blockIdx.x * blockDim.x + threadIdx.x;
    if (i < n) {
        float d = deg[i];
        deg[i] = (d > 0.0f) ? rsqrtf(d) : 0.0f;   // in-place: deg -> dinv
    }
}

// ---------------- GEMM1: hpre = x @ W1  (N x 256) @ (256 x 128) ----------------
// One wave computes one 16x16 output tile with V_WMMA_F32_16X16X4_F32.
// Block = 256 threads = 8 waves = the 8 column tiles of H=128.
// Grid = N/16 row tiles (N divisible by 16).

__global__ void k_gemm1_wmma(const float* __restrict__ x, const float* __restrict__ W1,
                             float* __restrict__ hpre) {
    const int lane = threadIdx.x & 31;
    const int wave = threadIdx.x >> 5;      // column tile 0..7
    const int r0   = blockIdx.x * 16;
    const int c0   = wave * 16;
    const int m    = lane & 15;             // row (A) / col (B,C) within tile
    const int half = lane >> 4;             // 0 or 1

    v8f acc = {};
    const float* arow = x + (size_t)(r0 + m) * DIN;

    for (int k = 0; k < DIN; k += 4) {
        const int ka = k + half * 2;        // A: lane half selects K pair
        v2f a; a.x = arow[ka]; a.y = arow[ka + 1];
        v2f b; b.x = W1[(size_t)ka * HID + c0 + m];
               b.y = W1[(size_t)(ka + 1) * HID + c0 + m];
        acc = __builtin_amdgcn_wmma_f32_16x16x4_f32(
            false, a, false, b, (short)0, acc, false, false);
    }
    // C/D layout: lanes 0-15 -> M=v, lanes 16-31 -> M=v+8; N = lane&15
    float* o = hpre + (size_t)(r0 + half * 8) * HID + c0 + m;
#pragma unroll
    for (int v = 0; v < 8; ++v) o[(size_t)v * HID] = acc[v];
}

// ---------------- self-loop init (doubles as zero-init of accumulator) ----------------

__global__ void k_selfinit1(const float* __restrict__ dinv, const float* __restrict__ hpre,
                            float* __restrict__ agg1, int n) {
    int idx = blockIdx.x * blockDim.x + threadIdx.x;
    if (idx < n * HID) {
        int i = idx >> 7;                   // /128
        float w = dinv[i]; w *= w;          // self-loop norm = dinv[i]^2
        agg1[idx] = w * hpre[idx];
    }
}

// ---------------- edge scatter, layer 1: one wave per edge, 4 cols/lane ----------------

__global__ void k_scatter1(const long long* __restrict__ src, const long long* __restrict__ dst,
                           const float* __restrict__ dinv, const float* __restrict__ hpre,
                           float* __restrict__ agg1, int e) {
    const int lane = threadIdx.x & 31;
    const int ed   = blockIdx.x * 8 + (threadIdx.x >> 5);
    if (ed >= e) return;                     // wave-uniform
    const int s = (int)src[ed];
    const int d = (int)dst[ed];
    const float w = dinv[s] * dinv[d];
    const float4 hv = *(const float4*)(hpre + (size_t)s * HID + lane * 4);
    float* ap = agg1 + (size_t)d * HID + lane * 4;
    atomicAdd(ap + 0, w * hv.x);
    atomicAdd(ap + 1, w * hv.y);
    atomicAdd(ap + 2, w * hv.z);
    atomicAdd(ap + 3, w * hv.w);
}

// ---------------- bias + ReLU (writes h1 into the hpre buffer, reuse) ----------------

__global__ void k_relu_bias(const float* __restrict__ agg1, const float* __restrict__ b1,
                            float* __restrict__ h1, int n) {
    int idx = blockIdx.x * blockDim.x + threadIdx.x;
    if (idx < n * HID) h1[idx] = fmaxf(agg1[idx] + b1[idx & (HID - 1)], 0.0f);
}

// ---------------- GEMM2: out2 = h1 @ W2  (N x 128) @ (128 x 8) ----------------
// B padded to 16 columns with zeros (select, not branch -> EXEC stays all-ones at WMMA).

__global__ void k_gemm2_wmma(const float* __restrict__ h1, const float* __restrict__ W2,
                             float* __restrict__ out2, int numTiles) {
    const int lane = threadIdx.x & 31;
    const int tile = blockIdx.x * 8 + (threadIdx.x >> 5);
    if (tile >= numTiles) return;            // wave-uniform
    const int r0   = tile * 16;
    const int m    = lane & 15;
    const int half = lane >> 4;

    v8f acc = {};
    const float* arow = h1 + (size_t)(r0 + m) * HID;

    for (int k = 0; k < HID; k += 4) {
        const int ka = k + half * 2;
        v2f a; a.x = arow[ka]; a.y = arow[ka + 1];
        v2f b;
        b.x = (m < NC) ? W2[(size_t)ka * NC + m] : 0.0f;
        b.y = (m < NC) ? W2[(size_t)(ka + 1) * NC + m] : 0.0f;
        acc = __builtin_amdgcn_wmma_f32_16x16x4_f32(
            false, a, false, b, (short)0, acc, false, false);
    }
    if (m < NC) {
        float* o = out2 + (size_t)(r0 + half * 8) * NC + m;
#pragma unroll
        for (int v = 0; v < 8; ++v) o[(size_t)v * NC] = acc[v];
    }
}

__global__ void k_selfinit2(const float* __restrict__ dinv, const float* __restrict__ out2,
                            float* __restrict__ agg2, int n) {
    int idx = blockIdx.x * blockDim.x + threadIdx.x;
    if (idx < n * NC) {
        int i = idx >> 3;
        float w = dinv[i]; w *= w;
        agg2[idx] = w * out2[idx];
    }
}

// ---------------- edge scatter, layer 2: thread per (edge, class) ----------------

__global__ void k_scatter2(const long long* __restrict__ src, const long long* __restrict__ dst,
                           const float* __restrict__ dinv, const float* __restrict__ out2,
                           float* __restrict__ agg2, int e) {
    int idx = blockIdx.x * blockDim.x + threadIdx.x;
    int ed = idx >> 3;
    int c  = idx & 7;
    if (ed >= e) return;
    int s = (int)src[ed];
    int d = (int)dst[ed];
    atomicAdd(agg2 + (size_t)d * NC + c, dinv[s] * dinv[d] * out2[(size_t)s * NC + c]);
}

// ---------------- bias + log_softmax over 8 classes ----------------

__global__ void k_logsoftmax(const float* __restrict__ agg2, const float* __restrict__ b2,
                             float* __restrict__ out, int n) {
    int i = blockIdx.x * blockDim.x + threadIdx.x;
    if (i >= n) return;
    float z[NC];
    float mx = -INFINITY;
#pragma unroll
    for (int c = 0; c < NC; ++c) {
        z[c] = agg2[(size_t)i * NC + c] + b2[c];
        mx = fmaxf(mx, z[c]);
    }
    float s = 0.0f;
#pragma unroll
    for (int c = 0; c < NC; ++c) s += expf(z[c] - mx);
    float l = logf(s);
#pragma unroll
    for (int c = 0; c < NC; ++c) out[(size_t)i * NC + c] = z[c] - mx - l;
}

// ---------------- launcher ----------------

extern "C" void kernel_launch(void* const* d_in, const int* in_sizes, int n_in,
                              void* d_out, int out_size, void* d_ws, size_t ws_size,
                              hipStream_t stream) {
    const float*     x  = (const float*)d_in[0];
    const float*     W1 = (const float*)d_in[1];
    const float*     b1 = (const float*)d_in[2];
    const float*     W2 = (const float*)d_in[3];
    const float*     b2 = (const float*)d_in[4];
    const long long* ei = (const long long*)d_in[5];   // int64 [2, E]

    const int N = in_sizes[0] / DIN;
    const int E = in_sizes[5] / 2;
    const long long* srcI = ei;
    const long long* dstI = ei + E;

    float* ws   = (float*)d_ws;
    float* dinv = ws;                           // N
    float* hpre = dinv + N;                     // N*128 (reused as h1)
    float* agg1 = hpre + (size_t)N * HID;       // N*128
    float* out2 = agg1 + (size_t)N * HID;       // N*8
    float* agg2 = out2 + (size_t)N * NC;        // N*8

    // normalization
    k_deg_init <<<(N + 255) / 256, 256, 0, stream>>>(dinv, N);
    k_deg_count<<<(E + 255) / 256, 256, 0, stream>>>(dstI, dinv, E);
    k_dinv     <<<(N + 255) / 256, 256, 0, stream>>>(dinv, N);

    // layer 1
    k_gemm1_wmma<<<N / 16, 256, 0, stream>>>(x, W1, hpre);
    k_selfinit1 <<<(N * HID + 255) / 256, 256, 0, stream>>>(dinv, hpre, agg1, N);
    k_scatter1  <<<(E + 7) / 8, 256, 0, stream>>>(srcI, dstI, dinv, hpre, agg1, E);
    k_relu_bias <<<(N * HID + 255) / 256, 256, 0, stream>>>(agg1, b1, hpre, N);

    // layer 2
    const int tiles2 = N / 16;
    k_gemm2_wmma<<<(tiles2 + 7) / 8, 256, 0, stream>>>(hpre, W2, out2, tiles2);
    k_selfinit2 <<<(N * NC + 255) / 256, 256, 0, stream>>>(dinv, out2, agg2, N);
    k_scatter2  <<<(E * NC + 255) / 256, 256, 0, stream>>>(srcI, dstI, dinv, out2, agg2, E);

    // output
    k_logsoftmax<<<(N + 255) / 256, 256, 0, stream>>>(agg2, b2, (float*)d_out, N);
}